// Net_41558103556103
// MI455X (gfx1250) — compile-verified
//
#include <hip/hip_runtime.h>

typedef __attribute__((ext_vector_type(2))) float v2f;
typedef __attribute__((ext_vector_type(8))) float v8f;

#define HID 20
#define KC  5        // 5 K-chunks of 4 cover K=20 exactly
#define HSTR 35      // LDS row stride: >=32 so cols 16..31 store in-row (branch-free),
                     // odd -> conflict-free row reads, 8*35%64=24 -> conflict-free stores

__global__ __launch_bounds__(256) void hyper_fused_kernel(
    const float* __restrict__ xf,    // [N]    inputs_for_f
    const float2* __restrict__ xg,   // [N,2]  inputs_for_g
    const float* __restrict__ W1,    // [20,1]
    const float* __restrict__ b1,    // [20]
    const float* __restrict__ W2,    // [20,20] row-major W2[n][k] = W2[n*20+k]
    const float* __restrict__ b2,    // [20]
    const float* __restrict__ Wg,    // [2,20]
    const float* __restrict__ bg,    // [2]
    float* __restrict__ out,         // [N]
    int n)
{
    // per-wave staging buffer for h2 (pre-relu): 32 samples x HSTR floats
    __shared__ float h2buf[8][32 * HSTR];

    const int lane = threadIdx.x & 31;
    const int wid  = threadIdx.x >> 5;
    const int col  = lane & 15;     // N index (B/C/D) and M index (A)
    const int hi   = lane >> 4;     // which half of the wave
    const int kh   = hi * 2;        // A/B k-offset within a 4-chunk: {0,2}

    // ---- hoisted, loop-invariant operand setup ----
    // layer-1 coefficients in the A-matrix VGPR layout
    float w1a[KC], w1b[KC], b1a[KC], b1b[KC];
#pragma unroll
    for (int kc = 0; kc < KC; ++kc) {
        int k0 = kc * 4 + kh;
        w1a[kc] = W1[k0];     w1b[kc] = W1[k0 + 1];
        b1a[kc] = b1[k0];     b1b[kc] = b1[k0 + 1];
    }
    // B operands for layer-2: B[k][nn] = W2[nn][k]; pad nn>=20 with zeros
    v2f Bt[2][KC];
#pragma unroll
    for (int t = 0; t < 2; ++t) {
        int nn = col + 16 * t;
#pragma unroll
        for (int kc = 0; kc < KC; ++kc) {
            int k0 = kc * 4 + kh;
            float bx = 0.f, by = 0.f;
            if (nn < HID) { bx = W2[nn * HID + k0]; by = W2[nn * HID + k0 + 1]; }
            Bt[t][kc].x = bx;
            Bt[t][kc].y = by;
        }
    }
    // bias b2 broadcast into accumulator init (C layout: value depends on N only)
    const float cinit0 = b2[col];
    const float cinit1 = (col + 16 < HID) ? b2[col + 16] : 0.f;
    // layer-3 weights (uniform indices -> scalar loads)
    float wg0[HID], wg1[HID];
#pragma unroll
    for (int j = 0; j < HID; ++j) { wg0[j] = Wg[j]; wg1[j] = Wg[HID + j]; }
    const float bg0 = bg[0], bg1 = bg[1];

    const int nChunks = (n + 31) >> 5;                 // 32 samples per wave-chunk
    const int waveId  = (blockIdx.x * blockDim.x + threadIdx.x) >> 5;
    const int nWaves  = (gridDim.x * blockDim.x) >> 5;
    const int nClamp  = n - 1;

    float* const hb = h2buf[wid];

    for (int chunk = waveId; chunk < nChunks; chunk += nWaves) {
        const int base = chunk << 5;

        v8f acc00, acc01, acc10, acc11;   // [sample-subtile][N-tile]
#pragma unroll
        for (int v = 0; v < 8; ++v) {
            acc00[v] = cinit0; acc01[v] = cinit1;
            acc10[v] = cinit0; acc11[v] = cinit1;
        }

        // x for the two 16-sample A-tiles (lane's A rows are M = col).
        // Clamped (not branched) loads keep the hot loop branch-free, EXEC all-ones.
        const float x0 = xf[min(base + col,      nClamp)];
        const float x1 = xf[min(base + 16 + col, nClamp)];

#pragma unroll
        for (int kc = 0; kc < KC; ++kc) {
            // layer-1 fused directly into A-operand layout (relu in VALU;
            // fma results are canonical so fmaxf lowers to a single dual max)
            v2f a0, a1;
            a0.x = fmaxf(fmaf(x0, w1a[kc], b1a[kc]), 0.f);
            a0.y = fmaxf(fmaf(x0, w1b[kc], b1b[kc]), 0.f);
            a1.x = fmaxf(fmaf(x1, w1a[kc], b1a[kc]), 0.f);
            a1.y = fmaxf(fmaf(x1, w1b[kc], b1b[kc]), 0.f);
            acc00 = __builtin_amdgcn_wmma_f32_16x16x4_f32(false, a0, false, Bt[0][kc], (short)0, acc00, false, false);
            acc01 = __builtin_amdgcn_wmma_f32_16x16x4_f32(false, a0, false, Bt[1][kc], (short)0, acc01, false, false);
            acc10 = __builtin_amdgcn_wmma_f32_16x16x4_f32(false, a1, false, Bt[0][kc], (short)0, acc10, false, false);
            acc11 = __builtin_amdgcn_wmma_f32_16x16x4_f32(false, a1, false, Bt[1][kc], (short)0, acc11, false, false);
        }

        // h2 (pre-relu) -> LDS: stores issue straight off the WMMA results; the
        // relu moves to the read side, which touches only 20 of the 32 stored
        // values per lane (fewer canonicalize+max pairs overall).
        // C layout: lane holds rows v + 8*hi, column col / col+16. Stores are
        // unconditional: cols 20..31 carry zeros (zero-padded B, zero bias) into
        // in-row slack words the read phase never touches.
#pragma unroll
        for (int v = 0; v < 8; ++v) {
            int r0 = v + 8 * hi;
            int r1 = 16 + v + 8 * hi;
            hb[r0 * HSTR + col]      = acc00[v];
            hb[r1 * HSTR + col]      = acc10[v];
            hb[r0 * HSTR + 16 + col] = acc01[v];
            hb[r1 * HSTR + 16 + col] = acc11[v];
        }
        // same-wave LDS ops are in-order (DScnt): a compiler scheduling barrier
        // is sufficient -- no cross-wave s_barrier needed (private LDS slice)
        __builtin_amdgcn_wave_barrier();

        // fused relu + layer-3 + per-sample dot: lane handles sample base+lane
        const int idx = base + lane;
        const float2 g = xg[min(idx, nClamp)];
        float r = fmaf(bg0, g.x, bg1 * g.y);
#pragma unroll
        for (int j = 0; j < HID; ++j) {
            float h = fmaxf(hb[lane * HSTR + j], 0.f);   // relu applied at read
            r = fmaf(h, fmaf(wg0[j], g.x, wg1[j] * g.y), r);
        }
        if (idx < n) out[idx] = r;
        __builtin_amdgcn_wave_barrier();   // reads done before next iter's stores
    }
}

extern "C" void kernel_launch(void* const* d_in, const int* in_sizes, int n_in,
                              void* d_out, int out_size, void* d_ws, size_t ws_size,
                              hipStream_t stream) {
    const float*  xf = (const float*)d_in[0];
    const float2* xg = (const float2*)d_in[1];
    const float*  W1 = (const float*)d_in[2];
    const float*  b1 = (const float*)d_in[3];
    const float*  W2 = (const float*)d_in[4];
    const float*  b2 = (const float*)d_in[5];
    const float*  Wg = (const float*)d_in[6];
    const float*  bg = (const float*)d_in[7];
    float* out = (float*)d_out;
    const int n = in_sizes[0];

    int blocks = 2048;                       // grid-stride; amortizes operand setup
    int maxBlocks = (n + 255) / 256;
    if (blocks > maxBlocks) blocks = maxBlocks;
    if (blocks < 1) blocks = 1;

    hyper_fused_kernel<<<blocks, 256, 0, stream>>>(xf, xg, W1, b1, W2, b2, Wg, bg, out, n);
}